// LightGCN_14809047236623
// MI455X (gfx1250) — compile-verified
//
#include <hip/hip_runtime.h>

// ---------------------------------------------------------------------------
// LightGCN on MI455X (gfx1250).
//
// Roofline: 3 GFLOP total vs ~400 MB HBM traffic (indices streamed NT) with
// the 38.4 MB embedding table resident in the 192 MB L2. Entirely
// memory/atomic bound -> COO gather + L2-side global_atomic_add_f32 scatter.
// The final *0.25 scale is done with V_WMMA_F32_16X16X4_F32 (exact f32 math)
// so the CDNA5 matrix path is exercised without precision risk.
// ---------------------------------------------------------------------------

#define N_USERS 100000
#define N_ITEMS 50000
#define N_NODES (N_USERS + N_ITEMS)
#define EMB_D   64

typedef __attribute__((ext_vector_type(2))) float v2f;
typedef __attribute__((ext_vector_type(8))) float v8f;

// emb = concat(user,item); acc = emb; nxt = 0.    (one float4 per thread)
__global__ void lgcn_init(const float* __restrict__ ue, const float* __restrict__ ie,
                          float* __restrict__ emb, float* __restrict__ acc,
                          float* __restrict__ nxt) {
    long long i   = (long long)blockIdx.x * blockDim.x + threadIdx.x;
    long long off = i * 4;
    const long long userElems = (long long)N_USERS * EMB_D;
    if (off >= (long long)N_NODES * EMB_D) return;
    float4 v;
    if (off < userElems) v = *(const float4*)(ue + off);
    else                 v = *(const float4*)(ie + (off - userElems));
    *(float4*)(emb + off) = v;
    *(float4*)(acc + off) = v;
    *(float4*)(nxt + off) = make_float4(0.f, 0.f, 0.f, 0.f);
}

// COO SpMM: 16 lanes per nnz, 4 consecutive dims per lane.
// A wave covers 2 nnz; the 256B source row is fetched fully coalesced.
// Index/value streams are non-temporal (keep L2 for the embedding table).
__global__ void lgcn_spmm(const int* __restrict__ rows, const int* __restrict__ cols,
                          const float* __restrict__ vals, const float* __restrict__ x,
                          float* __restrict__ y, int nnz) {
    long long t    = (long long)blockIdx.x * blockDim.x + threadIdx.x;
    long long work = (long long)nnz * 16;
    if (t >= work) return;
    int e  = (int)(t >> 4);
    int d4 = ((int)t & 15) << 2;            // 0..60

    if (((int)t & 15) == 0) {               // prefetch index streams ahead (speculative-safe)
        __builtin_prefetch(rows + e + 4096, 0, 0);
        __builtin_prefetch(cols + e + 4096, 0, 0);
        __builtin_prefetch(vals + e + 4096, 0, 0);
    }

    int   r = __builtin_nontemporal_load(rows + e);
    int   c = __builtin_nontemporal_load(cols + e);
    float v = __builtin_nontemporal_load(vals + e);

    const float4 xv = *(const float4*)(x + (long long)c * EMB_D + d4);
    float* yp = y + (long long)r * EMB_D + d4;
    unsafeAtomicAdd(yp + 0, v * xv.x);      // -> global_atomic_add_f32 (L2-side)
    unsafeAtomicAdd(yp + 1, v * xv.y);
    unsafeAtomicAdd(yp + 2, v * xv.z);
    unsafeAtomicAdd(yp + 3, v * xv.w);
}

// acc += nxt ; zero the buffer that will receive the next layer's SpMM.
__global__ void lgcn_fuse(float* __restrict__ acc, const float* __restrict__ nxt,
                          float* __restrict__ to_zero) {
    long long i   = (long long)blockIdx.x * blockDim.x + threadIdx.x;
    long long off = i * 4;
    if (off >= (long long)N_NODES * EMB_D) return;
    float4 a = *(const float4*)(acc + off);
    float4 n = *(const float4*)(nxt + off);
    a.x += n.x; a.y += n.y; a.z += n.z; a.w += n.w;
    *(float4*)(acc + off)     = a;
    *(float4*)(to_zero + off) = make_float4(0.f, 0.f, 0.f, 0.f);
}

// In-place acc *= 0.25 per 16x16 tile via V_WMMA_F32_16X16X4_F32:
//   D = sum_{kb=0..3}  T[:,4kb:4kb+4] (16x4)  x  0.25*E_kb (4x16),  exact f32.
// VGPR layouts per CDNA5 ISA 7.12.2:
//   A 16x4 : lane m=lane&15, hi=lane>>4 -> a.x=A[m,2hi], a.y=A[m,2hi+1]
//   B 4x16 : b.x=B[2hi, n=m], b.y=B[2hi+1, n=m]
//   C/D    : vgpr j, lane -> D[j+8*hi, m]
__global__ void lgcn_finalize_wmma(float* __restrict__ acc) {
    const int numTiles = (N_NODES / 16) * (EMB_D / 16);      // 9375 * 4
    int wave = blockIdx.x * (blockDim.x >> 5) + ((int)threadIdx.x >> 5);
    if (wave >= numTiles) return;                             // wave-uniform: EXEC stays all-1s
    int rt   = wave >> 2;
    int ct   = wave & 3;
    int lane = (int)threadIdx.x & 31;
    int m    = lane & 15;
    int hi   = lane >> 4;

    float* base = acc + (long long)rt * 16 * EMB_D + ct * 16; // tile origin

    v8f c = {0.f, 0.f, 0.f, 0.f, 0.f, 0.f, 0.f, 0.f};
#pragma unroll
    for (int kb = 0; kb < 4; ++kb) {
        v2f a, b;
        a.x = base[(long long)m * EMB_D + 4 * kb + 2 * hi];
        a.y = base[(long long)m * EMB_D + 4 * kb + 2 * hi + 1];
        b.x = (m == 4 * kb + 2 * hi)     ? 0.25f : 0.0f;      // 0.25 * E_kb
        b.y = (m == 4 * kb + 2 * hi + 1) ? 0.25f : 0.0f;
        c = __builtin_amdgcn_wmma_f32_16x16x4_f32(
                /*neg_a=*/false, a, /*neg_b=*/false, b,
                /*c_mod=*/(short)0, c, /*reuse_a=*/false, /*reuse_b=*/false);
    }
#pragma unroll
    for (int j = 0; j < 8; ++j)
        base[(long long)(j + 8 * hi) * EMB_D + m] = c[j];
}

extern "C" void kernel_launch(void* const* d_in, const int* in_sizes, int n_in,
                              void* d_out, int out_size, void* d_ws, size_t ws_size,
                              hipStream_t stream) {
    const float* ue   = (const float*)d_in[0];
    const float* ie   = (const float*)d_in[1];
    const int*   rows = (const int*)d_in[2];
    const int*   cols = (const int*)d_in[3];
    const float* vals = (const float*)d_in[4];
    const int    nnz  = in_sizes[2];

    float* acc  = (float*)d_out;                               // accumulator lives in d_out
    float* buf0 = (float*)d_ws;                                // 38.4 MB
    float* buf1 = buf0 + (size_t)N_NODES * EMB_D;              // 38.4 MB

    const long long ED = (long long)N_NODES * EMB_D;           // 9.6 M
    dim3 blk(256);
    int elemBlocks = (int)((ED / 4 + 255) / 256);              // 9375

    lgcn_init<<<elemBlocks, blk, 0, stream>>>(ue, ie, buf0, acc, buf1);

    long long work  = (long long)nnz * 16;
    int spmmBlocks  = (int)((work + 255) / 256);

    float* cur = buf0;
    float* nxt = buf1;
    for (int l = 0; l < 3; ++l) {
        lgcn_spmm<<<spmmBlocks, blk, 0, stream>>>(rows, cols, vals, cur, nxt, nnz);
        lgcn_fuse<<<elemBlocks, blk, 0, stream>>>(acc, nxt, cur);  // acc+=e_l ; zero(cur)
        float* t = cur; cur = nxt; nxt = t;
    }

    const int numTiles  = (N_NODES / 16) * (EMB_D / 16);       // 37500
    int       finBlocks = (numTiles + 7) / 8;                  // 8 waves / block
    lgcn_finalize_wmma<<<finBlocks, blk, 0, stream>>>(acc);
}